// PrototypeInteractionBlock_70471823393270
// MI455X (gfx1250) — compile-verified
//
#include <hip/hip_runtime.h>
#include <math.h>

typedef float v2f __attribute__((ext_vector_type(2)));
typedef float v8f __attribute__((ext_vector_type(8)));

#define NB 8
#define NC 256
#define NH 80
#define NW 80
#define NN 6400
#define NP 32
#define LN_EPS 1e-5f

// ---------------- workspace layout (float offsets) ----------------
static constexpr size_t OFF_POS    = 0;                       // 256*6400
static constexpr size_t OFF_SPART  = OFF_POS    + 1638400;    // 1600*256
static constexpr size_t OFF_POOLED = OFF_SPART  + 409600;     // 2048
static constexpr size_t OFF_H      = OFF_POOLED + 2048;       // 512
static constexpr size_t OFF_DYN    = OFF_H      + 512;        // 65536
static constexpr size_t OFF_K      = OFF_DYN    + 65536;      // 65536
static constexpr size_t OFF_V      = OFF_K      + 65536;      // 65536
static constexpr size_t OFF_U      = OFF_V      + 65536;      // 13107200
static constexpr size_t OFF_DWPART = OFF_U      + 13107200;   // 4096
static constexpr size_t OFF_BN2    = OFF_DWPART + 4096;       // 512 (mean,inv)
static constexpr size_t OFF_ZPART  = OFF_BN2    + 512;        // 4096
static constexpr size_t OFF_BN     = OFF_ZPART  + 4096;       // 512 (mean,inv)

#define WMMA_F32(a, b, c) \
  __builtin_amdgcn_wmma_f32_16x16x4_f32(false, (a), false, (b), (short)0, (c), false, false)

// ---------------- bilinear resize of pos embed 64x64 -> 80x80 (half-pixel) ----
__global__ void k_pos(const float* __restrict__ pe, float* __restrict__ pos) {
  int idx = blockIdx.x * 256 + threadIdx.x;
  if (idx >= NC * NN) return;
  int c = idx / NN, r = idx % NN;
  int i = r / NW, j = r % NW;
  float sy = fminf(fmaxf(i * 0.8f - 0.1f, 0.f), 63.f);
  float sx = fminf(fmaxf(j * 0.8f - 0.1f, 0.f), 63.f);
  int y0 = (int)sy, x0 = (int)sx;
  int y1 = min(y0 + 1, 63), x1 = min(x0 + 1, 63);
  float fy = sy - (float)y0, fx = sx - (float)x0;
  const float* p = pe + (size_t)c * 4096;
  float v00 = p[y0 * 64 + x0], v01 = p[y0 * 64 + x1];
  float v10 = p[y1 * 64 + x0], v11 = p[y1 * 64 + x1];
  pos[idx] = (v00 * (1.f - fx) + v01 * fx) * (1.f - fy) +
             (v10 * (1.f - fx) + v11 * fx) * fy;
}

// ---------------- per-token LN -> pooled partial sums (block = 32 tokens) ----
__global__ void k_lnpool(const float* __restrict__ x, const float* __restrict__ pos,
                         float* __restrict__ spart) {
  __shared__ float sp[8][256];
  int blk = blockIdx.x;                 // b*200 + chunk
  int b = blk / 200, chunk = blk % 200;
  int tid = threadIdx.x, wv = tid >> 5, lane = tid & 31;
  int tok0 = chunk * 32;
  float acc[8];
#pragma unroll
  for (int q = 0; q < 8; ++q) acc[q] = 0.f;
  for (int t = 0; t < 4; ++t) {
    int tok = tok0 + wv + t * 8;
    float xv[8], s = 0.f, s2 = 0.f;
#pragma unroll
    for (int q = 0; q < 8; ++q) {
      int c = lane + q * 32;
      float v = x[((size_t)(b * NC + c)) * NN + tok] + pos[(size_t)c * NN + tok];
      xv[q] = v; s += v; s2 += v * v;
    }
#pragma unroll
    for (int off = 16; off > 0; off >>= 1) {
      s  += __shfl_xor(s,  off, 32);
      s2 += __shfl_xor(s2, off, 32);
    }
    float mu  = s * (1.f / 256.f);
    float var = s2 * (1.f / 256.f) - mu * mu;
    float inv = rsqrtf(var + LN_EPS);
#pragma unroll
    for (int q = 0; q < 8; ++q) acc[q] += inv * (xv[q] - mu);
  }
#pragma unroll
  for (int q = 0; q < 8; ++q) sp[wv][lane + q * 32] = acc[q];
  __syncthreads();
  int c = tid;
  float sum = 0.f;
#pragma unroll
  for (int w = 0; w < 8; ++w) sum += sp[w][c];
  spart[(size_t)blk * 256 + c] = sum;
}

// ---------------- reduce partials -> pooled (with ln affine) ----------------
__global__ void k_pooled(const float* __restrict__ spart, const float* __restrict__ ln_w,
                         const float* __restrict__ ln_b, float* __restrict__ pooled) {
  int idx = blockIdx.x * 256 + threadIdx.x;   // 0..2047
  int b = idx >> 8, c = idx & 255;
  float s = 0.f;
  for (int ch = 0; ch < 200; ++ch) s += spart[((size_t)(b * 200 + ch)) * 256 + c];
  pooled[idx] = ln_w[c] * (s * (1.f / 6400.f)) + ln_b[c];
}

// ---------------- SE hidden: h = relu(pooled @ w1^T) ------------------------
__global__ void k_h(const float* __restrict__ pooled, const float* __restrict__ w1,
                    float* __restrict__ h) {
  int b = blockIdx.x, j = threadIdx.x;
  const float* wr = w1 + (size_t)j * 256;
  const float* pr = pooled + (size_t)b * 256;
  float a = 0.f;
  for (int c = 0; c < 256; ++c) a += wr[c] * pr[c];
  h[b * 64 + j] = fmaxf(a, 0.f);
}

// ---------------- dyn = proto + alpha * (h @ w2^T) --------------------------
__global__ void k_dyn(const float* __restrict__ h, const float* __restrict__ w2,
                      const float* __restrict__ proto, const float* __restrict__ alpha,
                      float* __restrict__ dyn, float* __restrict__ dyn_out) {
  int idx = blockIdx.x * 256 + threadIdx.x;   // 65536
  int b = idx >> 13, r = idx & 8191;
  const float* wr = w2 + (size_t)r * 64;
  const float* hb = h + b * 64;
  float a = 0.f;
#pragma unroll 8
  for (int j = 0; j < 64; ++j) a += wr[j] * hb[j];
  float d = proto[r] + alpha[0] * a;
  dyn[idx] = d;
  dyn_out[idx] = d;
}

// ---------------- k = dyn@wk^T+bk, v = dyn@wv^T+bv --------------------------
__global__ void k_kv(const float* __restrict__ dyn,
                     const float* __restrict__ wk, const float* __restrict__ bk,
                     const float* __restrict__ wvw, const float* __restrict__ bv,
                     float* __restrict__ km, float* __restrict__ vm) {
  int idx = blockIdx.x * 256 + threadIdx.x;   // (b*32+p)*256 + d
  int d = idx & 255, bp = idx >> 8;
  const float* row = dyn + (size_t)bp * 256;
  const float* wkr = wk + (size_t)d * 256;
  const float* wvr = wvw + (size_t)d * 256;
  float ak = bk[d], av = bv[d];
  for (int c = 0; c < 256; ++c) {
    float t = row[c];
    ak += t * wkr[c];
    av += t * wvr[c];
  }
  km[idx] = ak;
  vm[idx] = av;
}

// ---------------- fused q-proj / attention / out-proj (WMMA fp32) -----------
// block = 16 tokens of one batch; 4 waves; each wave owns pairs of 16x16
// output tiles sharing one A-fragment (halves LDS traffic, 2 indep WMMA chains)
__global__ void __launch_bounds__(128) k_attn(
    const float* __restrict__ x,  const float* __restrict__ pos,
    const float* __restrict__ wq, const float* __restrict__ bq,
    const float* __restrict__ km, const float* __restrict__ vm,
    const float* __restrict__ wo, const float* __restrict__ bo,
    float* __restrict__ u) {
  __shared__ __align__(16) float s_x[16 * 258];   // xp tile, later attn-out tile
  __shared__ __align__(16) float s_q[16 * 258];   // q tile, later u tile
  __shared__ __align__(16) float s_l[16 * 32];    // logits -> attn weights

  const int tid  = threadIdx.x;
  const int b    = blockIdx.x / 400;
  const int tok0 = (blockIdx.x % 400) * 16;
  const int wv   = tid >> 5;
  const int lane = tid & 31;
  const int m16  = lane & 15;
  const int hi   = lane >> 4;
  const int kk   = hi * 2;

  // phase 0: load xp tile [token][channel], b128 global loads
  for (int i = tid; i < 1024; i += 128) {
    int c = i >> 2, mq = (i & 3) * 4;
    float4 xv = *(const float4*)(x + ((size_t)(b * NC + c)) * NN + tok0 + mq);
    float4 pv = *(const float4*)(pos + (size_t)c * NN + tok0 + mq);
    s_x[(mq + 0) * 258 + c] = xv.x + pv.x;
    s_x[(mq + 1) * 258 + c] = xv.y + pv.y;
    s_x[(mq + 2) * 258 + c] = xv.z + pv.z;
    s_x[(mq + 3) * 258 + c] = xv.w + pv.w;
  }
  __syncthreads();

  // phase 1: q = xp @ wq^T + bq   (16x256, K=256), 2 tiles per wave iteration
  for (int it = 0; it < 2; ++it) {
    const int nt0 = wv * 2 + it * 8;
    const int n0 = nt0 * 16 + m16, n1 = n0 + 16;
    v8f acc0 = {0.f, 0.f, 0.f, 0.f, 0.f, 0.f, 0.f, 0.f};
    v8f acc1 = acc0;
    const float* ap  = s_x + m16 * 258 + kk;          // A[m][k]
    const float* bp0 = wq + (size_t)n0 * 256 + kk;    // B[k][n] = wq[n][k]
    const float* bp1 = wq + (size_t)n1 * 256 + kk;
    __builtin_prefetch(bp0, 0, 3);
    __builtin_prefetch(bp1, 0, 3);
#pragma unroll 4
    for (int kc = 0; kc < 64; ++kc) {
      v2f a  = *(const v2f*)(ap + kc * 4);
      v2f b0 = *(const v2f*)(bp0 + kc * 4);
      v2f b1 = *(const v2f*)(bp1 + kc * 4);
      acc0 = WMMA_F32(a, b0, acc0);
      acc1 = WMMA_F32(a, b1, acc1);
    }
    const float bias0 = bq[n0], bias1 = bq[n1];
#pragma unroll
    for (int r = 0; r < 8; ++r) {
      s_q[(r + 8 * hi) * 258 + n0] = acc0[r] + bias0;
      s_q[(r + 8 * hi) * 258 + n1] = acc1[r] + bias1;
    }
  }
  __syncthreads();

  // phase 2: logits = q @ k^T * C^-0.5   (16x32, K=256)
  if (wv < 2) {
    v8f acc = {0.f, 0.f, 0.f, 0.f, 0.f, 0.f, 0.f, 0.f};
    const int p = wv * 16 + m16;
    const float* bp = km + ((size_t)(b * NP + p)) * 256 + kk;  // B[k][p] = k[p][k]
    const float* ap = s_q + m16 * 258 + kk;
#pragma unroll 4
    for (int kc = 0; kc < 64; ++kc) {
      v2f a  = *(const v2f*)(ap + kc * 4);
      v2f bb = *(const v2f*)(bp + kc * 4);
      acc = WMMA_F32(a, bb, acc);
    }
#pragma unroll
    for (int r = 0; r < 8; ++r) s_l[(r + 8 * hi) * 32 + p] = acc[r] * 0.0625f;
  }
  __syncthreads();

  // phase 3: softmax over P=32 with clamp(+-30)
  if (tid < 16) {
    float* row = s_l + tid * 32;
    float lv[32], mx = -1e30f;
#pragma unroll
    for (int p = 0; p < 32; ++p) {
      float v = fminf(fmaxf(row[p], -30.f), 30.f);
      lv[p] = v; mx = fmaxf(mx, v);
    }
    float ssum = 0.f;
#pragma unroll
    for (int p = 0; p < 32; ++p) { float e = __expf(lv[p] - mx); lv[p] = e; ssum += e; }
    float rs = 1.f / ssum;
#pragma unroll
    for (int p = 0; p < 32; ++p) row[p] = lv[p] * rs;
  }
  __syncthreads();

  // phase 4: out = attn @ v   (16x256, K=32) -> reuse s_x
  for (int it = 0; it < 2; ++it) {
    const int nt0 = wv * 2 + it * 8;
    const int n0 = nt0 * 16 + m16, n1 = n0 + 16;
    v8f acc0 = {0.f, 0.f, 0.f, 0.f, 0.f, 0.f, 0.f, 0.f};
    v8f acc1 = acc0;
    const float* ap = s_l + m16 * 32 + kk;
#pragma unroll
    for (int kc = 0; kc < 8; ++kc) {
      v2f a = *(const v2f*)(ap + kc * 4);
      const int kb = kc * 4 + kk;
      const float* vr0 = vm + ((size_t)(b * NP + kb)) * 256;   // B[k][n] = v[k][n]
      const float* vr1 = vr0 + 256;
      v2f b0, b1;
      b0[0] = vr0[n0]; b0[1] = vr1[n0];
      b1[0] = vr0[n1]; b1[1] = vr1[n1];
      acc0 = WMMA_F32(a, b0, acc0);
      acc1 = WMMA_F32(a, b1, acc1);
    }
#pragma unroll
    for (int r = 0; r < 8; ++r) {
      s_x[(r + 8 * hi) * 258 + n0] = acc0[r];
      s_x[(r + 8 * hi) * 258 + n1] = acc1[r];
    }
  }
  __syncthreads();

  // phase 5: u = out @ wo^T + bo   (16x256, K=256), 2 tiles per wave iteration
  for (int it = 0; it < 2; ++it) {
    const int nt0 = wv * 2 + it * 8;
    const int n0 = nt0 * 16 + m16, n1 = n0 + 16;
    v8f acc0 = {0.f, 0.f, 0.f, 0.f, 0.f, 0.f, 0.f, 0.f};
    v8f acc1 = acc0;
    const float* ap  = s_x + m16 * 258 + kk;
    const float* bp0 = wo + (size_t)n0 * 256 + kk;
    const float* bp1 = wo + (size_t)n1 * 256 + kk;
    __builtin_prefetch(bp0, 0, 3);
    __builtin_prefetch(bp1, 0, 3);
#pragma unroll 4
    for (int kc = 0; kc < 64; ++kc) {
      v2f a  = *(const v2f*)(ap + kc * 4);
      v2f b0 = *(const v2f*)(bp0 + kc * 4);
      v2f b1 = *(const v2f*)(bp1 + kc * 4);
      acc0 = WMMA_F32(a, b0, acc0);
      acc1 = WMMA_F32(a, b1, acc1);
    }
    const float bias0 = bo[n0], bias1 = bo[n1];
#pragma unroll
    for (int r = 0; r < 8; ++r) {
      s_q[(r + 8 * hi) * 258 + n0] = acc0[r] + bias0;
      s_q[(r + 8 * hi) * 258 + n1] = acc1[r] + bias1;
    }
  }
  __syncthreads();

  // phase 6: coalesced b128 write of u tile
  for (int i = tid; i < 1024; i += 128) {
    int d = i >> 2, mq = (i & 3) * 4;
    float4 v;
    v.x = s_q[(mq + 0) * 258 + d];
    v.y = s_q[(mq + 1) * 258 + d];
    v.z = s_q[(mq + 2) * 258 + d];
    v.w = s_q[(mq + 3) * 258 + d];
    *(float4*)(u + ((size_t)(b * NC + d)) * NN + tok0 + mq) = v;
  }
}

// ---------------- depthwise 3x3 from an LDS-staged plane --------------------
__device__ __forceinline__ float dw3x3_lds(const float* plane, const float w[9],
                                           int h, int w_) {
  float s = 0.f;
#pragma unroll
  for (int dy = -1; dy <= 1; ++dy) {
    int hh = h + dy;
    if ((unsigned)hh >= (unsigned)NH) continue;
#pragma unroll
    for (int dx = -1; dx <= 1; ++dx) {
      int ww = w_ + dx;
      if ((unsigned)ww >= (unsigned)NW) continue;
      s += plane[hh * NW + ww] * w[(dy + 1) * 3 + (dx + 1)];
    }
  }
  return s;
}

// ---------------- dw conv batch-stat partials (block = one (b,c) plane) -----
__global__ void k_dwstat(const float* __restrict__ u, const float* __restrict__ dww,
                         float* __restrict__ part) {
  __shared__ __align__(16) float pl[NN];
  __shared__ float ss[256], sq[256];
  int bc = blockIdx.x, c = bc & 255;
  const float* up = u + (size_t)bc * NN;
  for (int i = threadIdx.x; i < NN / 4; i += 256)
    *(float4*)&pl[i * 4] = *(const float4*)&up[i * 4];
  const float* w9 = dww + c * 9;
  float w[9];
#pragma unroll
  for (int i = 0; i < 9; ++i) w[i] = w9[i];
  __syncthreads();
  float s = 0.f, s2 = 0.f;
  for (int i = threadIdx.x; i < NN; i += 256) {
    float d = dw3x3_lds(pl, w, i / NW, i % NW);
    s += d; s2 += d * d;
  }
  ss[threadIdx.x] = s; sq[threadIdx.x] = s2;
  __syncthreads();
  for (int st = 128; st > 0; st >>= 1) {
    if (threadIdx.x < st) {
      ss[threadIdx.x] += ss[threadIdx.x + st];
      sq[threadIdx.x] += sq[threadIdx.x + st];
    }
    __syncthreads();
  }
  if (threadIdx.x == 0) { part[bc * 2] = ss[0]; part[bc * 2 + 1] = sq[0]; }
}

// ---------------- finalize batch stats: mean + rsqrt(var+eps) ---------------
__global__ void k_statfin(const float* __restrict__ part, float* __restrict__ mean,
                          float* __restrict__ inv) {
  int c = threadIdx.x;
  float s = 0.f, s2 = 0.f;
  for (int b = 0; b < NB; ++b) {
    s  += part[(b * 256 + c) * 2];
    s2 += part[(b * 256 + c) * 2 + 1];
  }
  float m = s * (1.f / 51200.f);
  float v = s2 * (1.f / 51200.f) - m * m;
  mean[c] = m;
  inv[c] = rsqrtf(v + LN_EPS);
}

// ---------------- z = residual + (u + gelu(bn2(dw)))*gamma; stat partials ---
__global__ void k_z(const float* __restrict__ u, const float* __restrict__ x,
                    const float* __restrict__ dww,
                    const float* __restrict__ m2, const float* __restrict__ i2,
                    const float* __restrict__ bn2w, const float* __restrict__ bn2b,
                    const float* __restrict__ gamma,
                    float* __restrict__ z, float* __restrict__ part) {
  __shared__ __align__(16) float pl[NN];
  __shared__ float ss[256], sq[256];
  int bc = blockIdx.x, c = bc & 255;
  const float* up = u + (size_t)bc * NN;
  const float* xp = x + (size_t)bc * NN;
  float* zp = z + (size_t)bc * NN;
  for (int i = threadIdx.x; i < NN / 4; i += 256)
    *(float4*)&pl[i * 4] = *(const float4*)&up[i * 4];
  const float* w9 = dww + c * 9;
  float w[9];
#pragma unroll
  for (int i = 0; i < 9; ++i) w[i] = w9[i];
  float mm = m2[c], ii = i2[c], gw = bn2w[c], gb = bn2b[c], gm = gamma[c];
  __syncthreads();
  float s = 0.f, s2 = 0.f;
  for (int i = threadIdx.x; i < NN; i += 256) {
    float d = dw3x3_lds(pl, w, i / NW, i % NW);
    float t = (d - mm) * ii * gw + gb;
    float g = 0.5f * t * (1.f + erff(t * 0.70710678118654752f));   // exact GELU
    float zz = xp[i] + (pl[i] + g) * gm;
    zp[i] = zz; s += zz; s2 += zz * zz;
  }
  ss[threadIdx.x] = s; sq[threadIdx.x] = s2;
  __syncthreads();
  for (int st = 128; st > 0; st >>= 1) {
    if (threadIdx.x < st) {
      ss[threadIdx.x] += ss[threadIdx.x + st];
      sq[threadIdx.x] += sq[threadIdx.x + st];
    }
    __syncthreads();
  }
  if (threadIdx.x == 0) { part[bc * 2] = ss[0]; part[bc * 2 + 1] = sq[0]; }
}

// ---------------- y = bn(z) in place (b128 streaming) -----------------------
__global__ void k_y(float* __restrict__ z, const float* __restrict__ mean,
                    const float* __restrict__ inv, const float* __restrict__ bnw,
                    const float* __restrict__ bnb) {
  size_t i4 = ((size_t)blockIdx.x * 256 + threadIdx.x) * 4;
  int c = (int)((i4 / NN) & 255);
  float sc = inv[c] * bnw[c];
  float sb = bnb[c] - mean[c] * sc;
  float4 v = *(float4*)(z + i4);
  v.x = v.x * sc + sb;
  v.y = v.y * sc + sb;
  v.z = v.z * sc + sb;
  v.w = v.w * sc + sb;
  *(float4*)(z + i4) = v;
}

// ---------------- launch ----------------------------------------------------
extern "C" void kernel_launch(void* const* d_in, const int* in_sizes, int n_in,
                              void* d_out, int out_size, void* d_ws, size_t ws_size,
                              hipStream_t stream) {
  const float* x     = (const float*)d_in[0];
  const float* proto = (const float*)d_in[1];
  const float* pe    = (const float*)d_in[2];
  const float* ln_w  = (const float*)d_in[3];
  const float* ln_b  = (const float*)d_in[4];
  const float* sew1  = (const float*)d_in[5];
  const float* sew2  = (const float*)d_in[6];
  const float* alpha = (const float*)d_in[7];
  const float* wq    = (const float*)d_in[8];
  const float* bq    = (const float*)d_in[9];
  const float* wk    = (const float*)d_in[10];
  const float* bk    = (const float*)d_in[11];
  const float* wvw   = (const float*)d_in[12];
  const float* bv    = (const float*)d_in[13];
  const float* wo    = (const float*)d_in[14];
  const float* bo    = (const float*)d_in[15];
  const float* bn_w  = (const float*)d_in[16];
  const float* bn_b  = (const float*)d_in[17];
  const float* dww   = (const float*)d_in[18];
  const float* bn2w  = (const float*)d_in[19];
  const float* bn2b  = (const float*)d_in[20];
  const float* gamma = (const float*)d_in[21];

  float* out = (float*)d_out;          // y [13107200] ++ dyn [65536]
  float* ws  = (float*)d_ws;

  float* pos    = ws + OFF_POS;
  float* spart  = ws + OFF_SPART;
  float* pooled = ws + OFF_POOLED;
  float* h      = ws + OFF_H;
  float* dyn    = ws + OFF_DYN;
  float* km     = ws + OFF_K;
  float* vm     = ws + OFF_V;
  float* u      = ws + OFF_U;
  float* dwpart = ws + OFF_DWPART;
  float* bn2m   = ws + OFF_BN2;
  float* bn2i   = ws + OFF_BN2 + 256;
  float* zpart  = ws + OFF_ZPART;
  float* bnm    = ws + OFF_BN;
  float* bni    = ws + OFF_BN + 256;

  k_pos    <<<6400, 256, 0, stream>>>(pe, pos);
  k_lnpool <<<1600, 256, 0, stream>>>(x, pos, spart);
  k_pooled <<<8,    256, 0, stream>>>(spart, ln_w, ln_b, pooled);
  k_h      <<<8,    64,  0, stream>>>(pooled, sew1, h);
  k_dyn    <<<256,  256, 0, stream>>>(h, sew2, proto, alpha, dyn, out + 13107200);
  k_kv     <<<256,  256, 0, stream>>>(dyn, wk, bk, wvw, bv, km, vm);
  k_attn   <<<3200, 128, 0, stream>>>(x, pos, wq, bq, km, vm, wo, bo, u);
  k_dwstat <<<2048, 256, 0, stream>>>(u, dww, dwpart);
  k_statfin<<<1,    256, 0, stream>>>(dwpart, bn2m, bn2i);
  k_z      <<<2048, 256, 0, stream>>>(u, x, dww, bn2m, bn2i, bn2w, bn2b, gamma, out, zpart);
  k_statfin<<<1,    256, 0, stream>>>(zpart, bnm, bni);
  k_y      <<<12800,256, 0, stream>>>(out, bnm, bni, bn_w, bn_b);
}